// PhysicsLoss_79748952752291
// MI455X (gfx1250) — compile-verified
//
#include <hip/hip_runtime.h>
#include <hip/hip_bf16.h>
#include <math.h>

// Problem constants (match reference)
#define NNODES 20000
#define KNBR   64
#define BOND_TARGET_F  1.5f
#define TARGET_ANGLE_F 1.91113553093f   // 109.5 deg in radians
#define EPS_F          1e-8f
#define PAIRS_PER_NODE 2016.0f          // K*(K-1)/2
#define WAVES_PER_BLOCK 8               // 256 threads = 8 wave32

typedef __attribute__((ext_vector_type(2))) float v2f;
typedef __attribute__((ext_vector_type(8))) float v8f;

// One wave32 per node. Lane (half = lane>>4, idx = lane&15) owns rows
// {idx, idx+16, idx+32, idx+48} of the normalized neighbor matrix and keeps
// components (x,y) on half==0 lanes, (z,0) on half==1 lanes -> this register
// is simultaneously the A-tile (16x4 f32) and B-tile (4x16 f32) layout for
// V_WMMA_F32_16X16X4_F32. The 64x64 Gram needs only 10 upper-triangle tiles.
__global__ __launch_bounds__(256) void physics_node_kernel(
    const float* __restrict__ coords,   // N*3 f32
    const int*   __restrict__ dst,      // N*K int32 neighbor ids
    float*       __restrict__ ws)       // [0..N): bond partials, [N..2N): angle partials
{
    const int lane = threadIdx.x & 31;
    const int wave = threadIdx.x >> 5;
    const int node = blockIdx.x * WAVES_PER_BLOCK + wave;   // grid sized exactly
    const int half = lane >> 4;
    const int idx  = lane & 15;

    const float cx = coords[3 * node + 0];
    const float cy = coords[3 * node + 1];
    const float cz = coords[3 * node + 2];

    v2f R[4];
    float bond_acc = 0.0f;

#pragma unroll
    for (int g = 0; g < 4; ++g) {
        const int r  = 16 * g + idx;
        const int nb = dst[(size_t)node * KNBR + r];
        const float dx = coords[3 * nb + 0] - cx;     // ba = coords[nbr] - center
        const float dy = coords[3 * nb + 1] - cy;
        const float dz = coords[3 * nb + 2] - cz;
        const float d2 = dx * dx + dy * dy + dz * dz;
        const float inv = 1.0f / (sqrtf(d2) + EPS_F); // norm + eps in denominator
        const float hx = dx * inv, hy = dy * inv, hz = dz * inv;
        R[g].x = (half == 0) ? hx : hz;               // K pad (comp 3) = 0
        R[g].y = (half == 0) ? hy : 0.0f;
        // bond: dist = sqrt(d2 + eps); count each edge exactly once (half==0 lanes)
        const float dist = sqrtf(d2 + EPS_F);
        const float t = dist - BOND_TARGET_F;
        bond_acc += (half == 0) ? t * t : 0.0f;
    }

    float angle_acc = 0.0f;
#pragma unroll
    for (int i = 0; i < 4; ++i) {
#pragma unroll
        for (int j = i; j < 4; ++j) {
            v8f acc = {};
            // D = A x B + 0 : cos tile (rows 16i.., cols 16j..)
            acc = __builtin_amdgcn_wmma_f32_16x16x4_f32(
                /*neg_a=*/false, R[i], /*neg_b=*/false, R[j],
                /*c_mod=*/(short)0, acc, /*reuse_a=*/false, /*reuse_b=*/false);
#pragma unroll
            for (int v = 0; v < 8; ++v) {
                const int krow = 16 * i + v + 8 * half;  // C/D layout: M = v (+8 for hi half)
                const int lcol = 16 * j + idx;           // N = idx
                float c = acc[v];
                c = fminf(fmaxf(c, -0.999f), 0.999f);
                const float a = acosf(c) - TARGET_ANGLE_F;
                angle_acc += (krow < lcol) ? a * a : 0.0f;  // strict upper triangle
            }
        }
    }

    // wave32 reduction (fixed pattern -> deterministic)
#pragma unroll
    for (int off = 16; off > 0; off >>= 1) {
        bond_acc  += __shfl_down(bond_acc,  off, 32);
        angle_acc += __shfl_down(angle_acc, off, 32);
    }
    if (lane == 0) {
        ws[node]          = bond_acc;
        ws[NNODES + node] = angle_acc;
    }
}

// Single-block deterministic reduction of per-node partials -> scalar loss.
__global__ __launch_bounds__(256) void reduce_kernel(
    const float* __restrict__ ws, float* __restrict__ out)
{
    __shared__ float sb[256];
    __shared__ float sa[256];
    const int tid = threadIdx.x;
    float b = 0.0f, a = 0.0f;
    for (int i = tid; i < NNODES; i += 256) {   // fixed per-thread order
        b += ws[i];
        a += ws[NNODES + i];
    }
    sb[tid] = b;
    sa[tid] = a;
    __syncthreads();
    for (int s = 128; s > 0; s >>= 1) {
        if (tid < s) {
            sb[tid] += sb[tid + s];
            sa[tid] += sa[tid + s];
        }
        __syncthreads();
    }
    if (tid == 0) {
        const float bond_mean  = sb[0] / ((float)NNODES * (float)KNBR);
        const float angle_mean = sa[0] / ((float)NNODES * PAIRS_PER_NODE);
        out[0] = bond_mean + angle_mean;
    }
}

extern "C" void kernel_launch(void* const* d_in, const int* in_sizes, int n_in,
                              void* d_out, int out_size, void* d_ws, size_t ws_size,
                              hipStream_t stream) {
    const float* coords = (const float*)d_in[0];                 // (N,3) f32
    const int*   edge   = (const int*)d_in[1];                   // (2, N*K) int
    const int*   dst    = edge + (size_t)NNODES * KNBR;          // row 1 = dst
    float*       ws     = (float*)d_ws;                          // 2*N floats
    float*       out    = (float*)d_out;

    const int blocks = NNODES / WAVES_PER_BLOCK;                 // 2500
    physics_node_kernel<<<blocks, 256, 0, stream>>>(coords, dst, ws);
    reduce_kernel<<<1, 256, 0, stream>>>(ws, out);
}